// SplitMLP_51213190037980
// MI455X (gfx1250) — compile-verified
//
#include <hip/hip_runtime.h>

typedef float v2f __attribute__((ext_vector_type(2)));
typedef float v8f __attribute__((ext_vector_type(8)));

// One wave32 processes 16 batches:
//   D(16x16) = A(16x64) x B(64x16, block-diagonal from W) via 16 chained
//   V_WMMA_F32_16X16X4_F32, where chunk c covers k = 4c..4c+3 (patch p'=c).
// A-chunk per ISA layout: lane (l16, half) holds im[rb*2+half, cb*2 .. cb*2+1]
// as a float2 -> one b64 load, which is also exactly splits[b, c, 2*half..2*half+1].
__global__ __launch_bounds__(256) void splitmlp_wmma_kernel(
    const float* __restrict__ score_map,  // (B, 1, 8, 8)
    const float* __restrict__ Wmat,       // (16, 4)
    const float* __restrict__ bias_vec,   // (16,)
    float* __restrict__ agg_out,          // (B, 16)
    float* __restrict__ splits_out,       // (B, 16, 4)
    int Btot)
{
    const int lane  = threadIdx.x & 31;
    const int wave  = threadIdx.x >> 5;
    const int gwave = blockIdx.x * (blockDim.x >> 5) + wave;
    const int base  = gwave * 16;           // first batch of this wave's tile
    if (base >= Btot) return;               // wave-uniform exit (EXEC all-1 for WMMA)

    const int l16  = lane & 15;             // A-row (batch) / B-col (patch) index
    const int half = lane >> 4;             // selects K pair {0,1} vs {2,3}

    // Per-lane W pair: (W[l16, 2*half], W[l16, 2*half+1]) -- one b64 load.
    const v2f w01  = *(const v2f*)(Wmat + l16 * 4 + half * 2);
    const float bv = bias_vec[l16];

    const float* img = score_map  + (size_t)(base + l16) * 64;
    float*       sp  = splits_out + (size_t)(base + l16) * 64;

    v8f acc = {0.f, 0.f, 0.f, 0.f, 0.f, 0.f, 0.f, 0.f};
    const v2f zero2 = {0.f, 0.f};

#pragma unroll
    for (int c = 0; c < 16; ++c) {
        const int rb = c >> 2;              // patch row-block
        const int cb = c & 3;               // patch col-block

        // A chunk: two adjacent pixels of row rb*2+half at cols cb*2, cb*2+1.
        v2f a = *(const v2f*)(img + (rb * 2 + half) * 8 + cb * 2);

        // Block-diagonal B chunk: only column N == c is nonzero (lanes c, c+16).
        v2f bm = (l16 == c) ? w01 : zero2;

        acc = __builtin_amdgcn_wmma_f32_16x16x4_f32(
            /*neg_a=*/false, a, /*neg_b=*/false, bm,
            /*c_mod=*/(short)0, acc, /*reuse_a=*/false, /*reuse_b=*/false);

        // splits[b, c, 2*half .. 2*half+1] = a  (same registers, b64 store)
        *(v2f*)(sp + c * 4 + half * 2) = a;
    }

    // D layout: VGPR v, lanes 0-15 -> (batch base+v, p=lane),
    //           lanes 16-31 -> (batch base+v+8, p=lane-16).
#pragma unroll
    for (int v = 0; v < 8; ++v) {
        float x = acc[v] + bv;
        float y = 100.0f / (1.0f + __expf(-x));   // 100 * sigmoid(x)
        agg_out[(size_t)(base + v + half * 8) * 16 + l16] = y;
    }
}

extern "C" void kernel_launch(void* const* d_in, const int* in_sizes, int n_in,
                              void* d_out, int out_size, void* d_ws, size_t ws_size,
                              hipStream_t stream) {
    const float* score_map = (const float*)d_in[0];  // B*64 elems
    const float* Wmat      = (const float*)d_in[1];  // 64 elems
    const float* bias_vec  = (const float*)d_in[2];  // 16 elems

    const int Btot = in_sizes[0] / 64;

    float* agg    = (float*)d_out;                       // B*16
    float* splits = (float*)d_out + (size_t)Btot * 16;   // B*64

    const int waves  = (Btot + 15) / 16;
    const int blocks = (waves + 7) / 8;   // 8 waves (256 threads) per block

    splitmlp_wmma_kernel<<<blocks, 256, 0, stream>>>(
        score_map, Wmat, bias_vec, agg, splits, Btot);
}